// MarketGNN_22299470201097
// MI455X (gfx1250) — compile-verified
//
#include <hip/hip_runtime.h>
#include <hip/hip_bf16.h>

// ---------------------------------------------------------------------------
// GraphSAGE 2-layer + linear head for MI455X (gfx1250, wave32).
// Edge phase: fp32 L2-resident atomics (node state << 192MB L2).
// Dense phase: V_WMMA_F32_16X16X4_F32, one 16-node tile per wave.
// ---------------------------------------------------------------------------

typedef __attribute__((ext_vector_type(2))) float v2f;
typedef __attribute__((ext_vector_type(8))) float v8f;

#define IN_DIM  32
#define HID_DIM 16
#define OUT_DIM 8

__device__ __forceinline__ v8f wmma_k4(v2f a, v2f b, v8f c) {
  // D = A(16x4) * B(4x16) + C(16x16), fp32.
  return __builtin_amdgcn_wmma_f32_16x16x4_f32(
      /*neg_a=*/false, a, /*neg_b=*/false, b,
      /*c_mod=*/(short)0, c, /*reuse_a=*/false, /*reuse_b=*/false);
}

// ---------------------------------------------------------------------------
// Zero the accumulator region of the workspace (deg + msg1 + msg2 = 49N f32).
// ---------------------------------------------------------------------------
__global__ void zero_f4(float4* __restrict__ p, int n4) {
  int t = blockIdx.x * blockDim.x + threadIdx.x;
  if (t < n4) p[t] = make_float4(0.f, 0.f, 0.f, 0.f);
}

// ---------------------------------------------------------------------------
// Layer-1 scatter: msg1[dst] += x[src] (32 dims), deg[dst] += 1.
// One thread per (edge, 4-dim chunk): E*8 threads, float4 gather + 4 atomics.
// ---------------------------------------------------------------------------
__global__ void scatter1(const float* __restrict__ x,
                         const int* __restrict__ src,
                         const int* __restrict__ dst,
                         float* __restrict__ msg1,
                         float* __restrict__ deg, int E) {
  int t = blockIdx.x * blockDim.x + threadIdx.x;
  if (t >= E * 8) return;
  int e = t >> 3;
  int q = (t & 7) << 2;            // dim offset 0,4,...,28
  int s = src[e];
  int d = dst[e];
  float4 v = *(const float4*)(x + s * IN_DIM + q);
  float* p = msg1 + d * IN_DIM + q;
  atomicAdd(p + 0, v.x);
  atomicAdd(p + 1, v.y);
  atomicAdd(p + 2, v.z);
  atomicAdd(p + 3, v.w);
  if (q == 0) atomicAdd(deg + d, 1.0f);
}

// ---------------------------------------------------------------------------
// Layer-2 scatter: msg2[dst] += h1[src] (16 dims). E*4 threads.
// ---------------------------------------------------------------------------
__global__ void scatter2(const float* __restrict__ h1,
                         const int* __restrict__ src,
                         const int* __restrict__ dst,
                         float* __restrict__ msg2, int E) {
  int t = blockIdx.x * blockDim.x + threadIdx.x;
  if (t >= E * 4) return;
  int e = t >> 2;
  int q = (t & 3) << 2;            // dim offset 0,4,8,12
  int s = src[e];
  int d = dst[e];
  float4 v = *(const float4*)(h1 + s * HID_DIM + q);
  float* p = msg2 + d * HID_DIM + q;
  atomicAdd(p + 0, v.x);
  atomicAdd(p + 1, v.y);
  atomicAdd(p + 2, v.z);
  atomicAdd(p + 3, v.w);
}

// ---------------------------------------------------------------------------
// Layer 1 dense: h1 = relu( (msg1/max(deg,1)) @ Wl1^T + bl1 + x @ Wr1^T )
// One wave per 16-node tile. K=32 -> 8 chained WMMA per matrix product.
//
// WMMA f32 16x16x4 register layouts (ISA 7.12.2):
//  A 16x4 : lane<16 holds row M=lane, V0=K0,V1=K1 ; lane>=16 same rows, K2/K3
//  B 4x16 : lane holds col N=lane&15 ; lanes<16: K0(V0),K1(V1); lanes>=16: K2,K3
//  D 16x16: VGPR v -> row v (lanes<16) / row v+8 (lanes>=16), col N=lane&15
// ---------------------------------------------------------------------------
__global__ __launch_bounds__(256) void layer1_dense(
    const float* __restrict__ x, const float* __restrict__ msg1,
    const float* __restrict__ deg,
    const float* __restrict__ Wl1, const float* __restrict__ bl1,
    const float* __restrict__ Wr1,
    float* __restrict__ h1, int ntiles) {
  int wave = blockIdx.x * (blockDim.x >> 5) + (threadIdx.x >> 5);
  if (wave >= ntiles) return;                 // whole-wave exit: EXEC stays all-1
  int lane = threadIdx.x & 31;
  int half = lane >> 4;                        // K half for A/B loads
  int j    = lane & 15;                        // col / A-row index
  int rowA = wave * 16 + j;

  float scale = 1.0f / fmaxf(deg[rowA], 1.0f);

  v8f accL = {};                               // mean @ Wl1^T
  v8f accR = {};                               // x    @ Wr1^T
#pragma unroll
  for (int kb = 0; kb < IN_DIM / 4; ++kb) {
    int k0 = kb * 4 + half * 2;
    const float* pm = msg1 + rowA * IN_DIM + k0;
    v2f a;  a.x = pm[0] * scale; a.y = pm[1] * scale;
    const float* pw = Wl1 + j * IN_DIM + k0;   // B[k][j] = Wl1[j][k]
    v2f b;  b.x = pw[0]; b.y = pw[1];
    accL = wmma_k4(a, b, accL);

    const float* px = x + rowA * IN_DIM + k0;
    v2f ar; ar.x = px[0]; ar.y = px[1];
    const float* pr = Wr1 + j * IN_DIM + k0;
    v2f br; br.x = pr[0]; br.y = pr[1];
    accR = wmma_k4(ar, br, accR);
  }

  float bj = bl1[j];
  int rbase = wave * 16 + half * 8;
#pragma unroll
  for (int v = 0; v < 8; ++v) {
    float val = accL[v] + accR[v] + bj;
    val = fmaxf(val, 0.0f);                    // ReLU
    h1[(rbase + v) * HID_DIM + j] = val;
  }
}

// ---------------------------------------------------------------------------
// Layer 2 dense + fused head:
//   h2 = (msg2/max(deg,1)) @ Wl2^T + bl2 + h1 @ Wr2^T      (16x8, zero-padded)
//   out = h2 @ Wfc^T + bfc                                  (16x1 per tile)
// K=16 -> 4 chained WMMA per product. Head reduced with 16-lane shfl_xor.
// ---------------------------------------------------------------------------
__global__ __launch_bounds__(256) void layer2_head(
    const float* __restrict__ h1, const float* __restrict__ msg2,
    const float* __restrict__ deg,
    const float* __restrict__ Wl2, const float* __restrict__ bl2,
    const float* __restrict__ Wr2,
    const float* __restrict__ Wfc, const float* __restrict__ bfc,
    float* __restrict__ out, int ntiles) {
  int wave = blockIdx.x * (blockDim.x >> 5) + (threadIdx.x >> 5);
  if (wave >= ntiles) return;
  int lane = threadIdx.x & 31;
  int half = lane >> 4;
  int j    = lane & 15;
  int rowA = wave * 16 + j;
  bool jok = (j < OUT_DIM);                    // cols 8..15 are zero padding

  float scale = 1.0f / fmaxf(deg[rowA], 1.0f);

  v8f accL = {};
  v8f accR = {};
#pragma unroll
  for (int kb = 0; kb < HID_DIM / 4; ++kb) {
    int k0 = kb * 4 + half * 2;
    const float* pm = msg2 + rowA * HID_DIM + k0;
    v2f a;  a.x = pm[0] * scale; a.y = pm[1] * scale;
    v2f b;  b.x = jok ? Wl2[j * HID_DIM + k0]     : 0.0f;
            b.y = jok ? Wl2[j * HID_DIM + k0 + 1] : 0.0f;
    accL = wmma_k4(a, b, accL);

    const float* ph = h1 + rowA * HID_DIM + k0;
    v2f ar; ar.x = ph[0]; ar.y = ph[1];
    v2f br; br.x = jok ? Wr2[j * HID_DIM + k0]     : 0.0f;
            br.y = jok ? Wr2[j * HID_DIM + k0 + 1] : 0.0f;
    accR = wmma_k4(ar, br, accR);
  }

  float bj = jok ? bl2[j] : 0.0f;
  float wf = jok ? Wfc[j] : 0.0f;
  float bias = bfc[0];
  int rbase = wave * 16 + half * 8;
#pragma unroll
  for (int v = 0; v < 8; ++v) {
    float h2v = accL[v] + accR[v] + bj;         // h2[row][j] (0 for j>=8)
    float val = h2v * wf;                       // head partial
    val += __shfl_xor(val, 1);                  // reduce within 16-lane half
    val += __shfl_xor(val, 2);
    val += __shfl_xor(val, 4);
    val += __shfl_xor(val, 8);
    if (j == 0) out[rbase + v] = val + bias;    // lanes 0 and 16 write
  }
}

// ---------------------------------------------------------------------------
// Launch. Inputs (setup_inputs order):
//  0:x[N*32] 1:edge_index[2*E] 2:Wl1[16*32] 3:bl1[16] 4:Wr1[16*32]
//  5:Wl2[8*16] 6:bl2[8] 7:Wr2[8*16] 8:Wfc[8] 9:bfc[1]
// Workspace layout (floats): deg[N] | msg1[32N] | msg2[16N] | h1[16N]
// ---------------------------------------------------------------------------
extern "C" void kernel_launch(void* const* d_in, const int* in_sizes, int n_in,
                              void* d_out, int out_size, void* d_ws, size_t ws_size,
                              hipStream_t stream) {
  const float* x   = (const float*)d_in[0];
  const int*   ei  = (const int*)d_in[1];
  const float* Wl1 = (const float*)d_in[2];
  const float* bl1 = (const float*)d_in[3];
  const float* Wr1 = (const float*)d_in[4];
  const float* Wl2 = (const float*)d_in[5];
  const float* bl2 = (const float*)d_in[6];
  const float* Wr2 = (const float*)d_in[7];
  const float* Wfc = (const float*)d_in[8];
  const float* bfc = (const float*)d_in[9];

  const int N = in_sizes[0] / IN_DIM;       // 100000
  const int E = in_sizes[1] / 2;            // 1600000
  const int* srcIdx = ei;                   // edge_index[0, :]
  const int* dstIdx = ei + E;               // edge_index[1, :]

  float* ws   = (float*)d_ws;
  float* deg  = ws;                         // [N]
  float* msg1 = ws + N;                     // [32N]
  float* msg2 = ws + 33 * N;                // [16N]
  float* h1   = ws + 49 * N;                // [16N]

  float* outp = (float*)d_out;

  const int B = 256;

  // 1) zero accumulators: deg + msg1 + msg2 = 49N floats (16B-aligned, 49N%4==0)
  int nz4 = (49 * N) / 4;
  zero_f4<<<(nz4 + B - 1) / B, B, 0, stream>>>((float4*)ws, nz4);

  // 2) layer-1 edge scatter (+degree)
  int t1 = E * 8;
  scatter1<<<(t1 + B - 1) / B, B, 0, stream>>>(x, srcIdx, dstIdx, msg1, deg, E);

  // 3) layer-1 dense (WMMA) + ReLU
  int ntiles = N / 16;                      // 6250
  int wavesPerBlk = B / 32;
  layer1_dense<<<(ntiles + wavesPerBlk - 1) / wavesPerBlk, B, 0, stream>>>(
      x, msg1, deg, Wl1, bl1, Wr1, h1, ntiles);

  // 4) layer-2 edge scatter
  int t2 = E * 4;
  scatter2<<<(t2 + B - 1) / B, B, 0, stream>>>(h1, srcIdx, dstIdx, msg2, E);

  // 5) layer-2 dense (WMMA) + fused linear head
  layer2_head<<<(ntiles + wavesPerBlk - 1) / wavesPerBlk, B, 0, stream>>>(
      h1, msg2, deg, Wl2, bl2, Wr2, Wfc, bfc, outp, ntiles);
}